// RPN_Deep_11974368821928
// MI455X (gfx1250) — compile-verified
//
#include <hip/hip_runtime.h>
#include <math.h>

// ---------------------------------------------------------------------------
// SiamRPN deformable head for MI455X (gfx1250, wave32).
// f32 end-to-end; GEMM stages use V_WMMA_F32_16X16X4_F32.
// ---------------------------------------------------------------------------

typedef __attribute__((ext_vector_type(2))) float v2f;
typedef __attribute__((ext_vector_type(8))) float v8f;

#define BN_EPS 1e-5f

static __device__ __forceinline__ v8f wmma4(v2f a, v2f b, v8f c) {
  // D = A(16x4) * B(4x16) + C(16x16), all f32
  return __builtin_amdgcn_wmma_f32_16x16x4_f32(false, a, false, b, (short)0, c,
                                               false, false);
}

// --- generic weight transpose: Bm[k*Npad + n] = (n<N) ? W[n*K + k] : 0 ------
__global__ void k_transpose(const float* __restrict__ W, float* __restrict__ Bm,
                            int N, int K, int Npad) {
  int idx = blockIdx.x * blockDim.x + threadIdx.x;
  if (idx >= K * Npad) return;
  int k = idx / Npad, n = idx - k * Npad;
  Bm[idx] = (n < N) ? W[n * K + k] : 0.0f;
}

// --- offset conv (3x3, 256 -> 27, padded to 32) via im2col WMMA -------------
// block = 32 threads (1 wave), grid = B * ceil(hw/16)
__global__ void k_offconv_wmma(const float* __restrict__ x,   // [B,256,Hin,Win]
                               const float* __restrict__ Bm,  // [2304][32]
                               const float* __restrict__ bias,// [27]
                               float* __restrict__ om,        // [B,27,hw]
                               int Hin, int Win, int ho, int wo) {
  const int hw = ho * wo;
  const int tiles = (hw + 15) >> 4;
  const int b = blockIdx.x / tiles;
  const int pix0 = (blockIdx.x - b * tiles) << 4;
  const int lane = threadIdx.x;
  const int half = lane >> 4, l16 = lane & 15;

  __shared__ float aLds[16 * 145];
  const float* xb = x + (size_t)b * 256 * Hin * Win;

  v8f acc0 = {0.f, 0.f, 0.f, 0.f, 0.f, 0.f, 0.f, 0.f};
  v8f acc1 = acc0;

  for (int cblk = 0; cblk < 16; ++cblk) {
    // stage A chunk: 16 positions x 144 (16 channels * 9 taps)
    for (int e = lane; e < 16 * 144; e += 32) {
      int mr = e / 144, kl = e - mr * 144;
      int cl = kl / 9, kk = kl - cl * 9;
      int c = (cblk << 4) + cl;
      int pix = pix0 + mr;
      float v = 0.0f;
      if (pix < hw) {
        int h = pix / wo, w = pix - h * wo;
        int ky = kk / 3, kx = kk - ky * 3;
        v = xb[((size_t)c * Hin + (h + ky)) * Win + (w + kx)];
      }
      aLds[mr * 145 + kl] = v;
    }
    __syncthreads();

    const float* Bk = Bm + (size_t)cblk * 144 * 32;
    for (int k0 = 0; k0 < 144; k0 += 4) {
      int kb = k0 + (half << 1);
      v2f a;
      a[0] = aLds[l16 * 145 + kb];
      a[1] = aLds[l16 * 145 + kb + 1];
      v2f b0, b1;
      b0[0] = Bk[kb * 32 + l16];
      b0[1] = Bk[(kb + 1) * 32 + l16];
      b1[0] = Bk[kb * 32 + 16 + l16];
      b1[1] = Bk[(kb + 1) * 32 + 16 + l16];
      acc0 = wmma4(a, b0, acc0);
      acc1 = wmma4(a, b1, acc1);
    }
    __syncthreads();
  }

  // epilogue (+bias), keep only n < 27
#pragma unroll
  for (int r = 0; r < 8; ++r) {
    int pix = pix0 + r + (half << 3);
    if (pix >= hw) continue;
    int n0 = l16;
    om[((size_t)b * 27 + n0) * hw + pix] = acc0[r] + bias[n0];
    int n1 = 16 + l16;
    if (n1 < 27) om[((size_t)b * 27 + n1) * hw + pix] = acc1[r] + bias[n1];
  }
}

// --- bilinear tap precompute: clamp addresses, fold validity*mask into wts --
__global__ void k_interp(const float* __restrict__ om,  // [B,27,hw]
                         int4* __restrict__ toff,       // [B*9*hw]
                         float4* __restrict__ twt,      // [B*9*hw]
                         int Hin, int Win, int ho, int wo) {
  const int hw = ho * wo;
  int idx = blockIdx.x * blockDim.x + threadIdx.x;
  if (idx >= 16 * 9 * hw) return;
  int pix = idx % hw;
  int k = (idx / hw) % 9;
  int b = idx / (hw * 9);
  const float* omb = om + (size_t)b * 27 * hw;
  float oy = omb[(size_t)(2 * k) * hw + pix];
  float ox = omb[(size_t)(2 * k + 1) * hw + pix];
  float mv = omb[(size_t)(18 + k) * hw + pix];
  float msk = 1.0f / (1.0f + expf(-mv));

  int h = pix / wo, w = pix - h * wo;
  int ky = k / 3, kx = k - ky * 3;
  float py = (float)(ky + h) + oy;
  float px = (float)(kx + w) + ox;
  float fy0 = floorf(py), fx0 = floorf(px);
  float dy = py - fy0, dx = px - fx0;
  int y0 = (int)fy0, x0 = (int)fx0;

  float wq[4] = {(1.f - dy) * (1.f - dx), (1.f - dy) * dx,
                 dy * (1.f - dx), dy * dx};
  int cy[4] = {y0, y0, y0 + 1, y0 + 1};
  int cx[4] = {x0, x0 + 1, x0, x0 + 1};
  int o[4];
  float wv[4];
#pragma unroll
  for (int q = 0; q < 4; ++q) {
    bool valid = (cy[q] >= 0) && (cy[q] < Hin) && (cx[q] >= 0) && (cx[q] < Win);
    int yc = min(max(cy[q], 0), Hin - 1);
    int xc = min(max(cx[q], 0), Win - 1);
    o[q] = yc * Win + xc;
    wv[q] = valid ? wq[q] * msk : 0.0f;
  }
  toff[idx] = make_int4(o[0], o[1], o[2], o[3]);
  twt[idx] = make_float4(wv[0], wv[1], wv[2], wv[3]);
}

// --- deformable main conv GEMM (M=16 tile, N=256, K=2304) + bias + BN -------
// block = 128 threads (4 waves x N=64), grid = B * ceil(hw/16)
__global__ void k_dcn_wmma(const float* __restrict__ x,       // [B,256,Hin,Win]
                           const int4* __restrict__ toff,     // [B*9*hw]
                           const float4* __restrict__ twt,    // [B*9*hw]
                           const float* __restrict__ Bm,      // [2304][256]
                           const float* __restrict__ bias,    // [256]
                           const float* __restrict__ g, const float* __restrict__ bb,
                           const float* __restrict__ mn, const float* __restrict__ vr,
                           float* __restrict__ out,           // [B,256,hw]
                           int Hin, int Win, int ho, int wo) {
  const int hw = ho * wo;
  const int tiles = (hw + 15) >> 4;
  const int b = blockIdx.x / tiles;
  const int pix0 = (blockIdx.x - b * tiles) << 4;
  const int tid = threadIdx.x;
  const int lane = tid & 31, wv = tid >> 5;
  const int half = lane >> 4, l16 = lane & 15;

  __shared__ float aLds[16 * 145];
  __shared__ int tO[144][4];
  __shared__ float tW[144][4];

  // load 16 positions x 9 taps of interpolation metadata
  for (int e = tid; e < 144; e += 128) {
    int mr = e / 9, kk = e - mr * 9;
    int pix = pix0 + mr;
    int4 o = make_int4(0, 0, 0, 0);
    float4 wt = make_float4(0.f, 0.f, 0.f, 0.f);
    if (pix < hw) {
      size_t t = ((size_t)b * 9 + kk) * hw + pix;
      o = toff[t];
      wt = twt[t];
    }
    tO[e][0] = o.x; tO[e][1] = o.y; tO[e][2] = o.z; tO[e][3] = o.w;
    tW[e][0] = wt.x; tW[e][1] = wt.y; tW[e][2] = wt.z; tW[e][3] = wt.w;
  }
  __syncthreads();

  v8f acc[4];
#pragma unroll
  for (int j = 0; j < 4; ++j) acc[j] = (v8f){0.f,0.f,0.f,0.f,0.f,0.f,0.f,0.f};

  const float* xb = x + (size_t)b * 256 * Hin * Win;
  const int nbase = wv * 64;

  for (int cblk = 0; cblk < 16; ++cblk) {
    // stage A chunk: gathered bilinear samples, 16 x 144
    for (int e = tid; e < 16 * 144; e += 128) {
      int mr = e / 144, kl = e - mr * 144;
      int cl = kl / 9, kk = kl - cl * 9;
      int c = (cblk << 4) + cl;
      const float* xc = xb + (size_t)c * Hin * Win;
      int ti = mr * 9 + kk;
      float v = tW[ti][0] * xc[tO[ti][0]] + tW[ti][1] * xc[tO[ti][1]] +
                tW[ti][2] * xc[tO[ti][2]] + tW[ti][3] * xc[tO[ti][3]];
      aLds[mr * 145 + kl] = v;
    }
    __syncthreads();

    const float* Bk = Bm + (size_t)(cblk * 144) * 256;
    for (int k0 = 0; k0 < 144; k0 += 4) {
      int kb = k0 + (half << 1);
      v2f a;
      a[0] = aLds[l16 * 145 + kb];
      a[1] = aLds[l16 * 145 + kb + 1];
      const float* Br0 = Bk + (size_t)kb * 256 + nbase + l16;
      const float* Br1 = Br0 + 256;
#pragma unroll
      for (int j = 0; j < 4; ++j) {
        v2f bf;
        bf[0] = Br0[j * 16];
        bf[1] = Br1[j * 16];
        acc[j] = wmma4(a, bf, acc[j]);
      }
    }
    __syncthreads();
  }

  // epilogue: +bias, BN
#pragma unroll
  for (int j = 0; j < 4; ++j) {
    int n = nbase + j * 16 + l16;
    float scale = g[n] * rsqrtf(vr[n] + BN_EPS);
    float shift = bb[n] - mn[n] * scale;
    float bs = bias[n];
#pragma unroll
    for (int r = 0; r < 8; ++r) {
      int pix = pix0 + r + (half << 3);
      if (pix < hw)
        out[((size_t)b * 256 + n) * hw + pix] = (acc[j][r] + bs) * scale + shift;
    }
  }
}

// --- depthwise cross-correlation: inst 30x30 (x) exam 18x18 -> 13x13 --------
__global__ void k_xcorr(const float* __restrict__ inst,  // [B*256, 900]
                        const float* __restrict__ exam,  // [B*256, 324]
                        float* __restrict__ out) {       // [B*256, 169]
  __shared__ float eL[324];
  __shared__ float iL[900];
  const int bc = blockIdx.x;
  const int tid = threadIdx.x;
  const float* e = exam + (size_t)bc * 324;
  const float* in = inst + (size_t)bc * 900;
  for (int j = tid; j < 324; j += 256) eL[j] = e[j];
  for (int j = tid; j < 900; j += 256) iL[j] = in[j];
  __syncthreads();
  if (tid < 169) {
    int u = tid / 13, v = tid - u * 13;
    float acc = 0.f;
    for (int y = 0; y < 18; ++y) {
      const float* ir = &iL[(u + y) * 30 + v];
      const float* er = &eL[y * 18];
#pragma unroll
      for (int xx = 0; xx < 18; ++xx) acc = fmaf(ir[xx], er[xx], acc);
    }
    out[(size_t)bc * 169 + tid] = acc;
  }
}

// --- 1x1 fuse GEMM (M tiles of 16 over 169 px, N=256, K=256) + BN + ReLU ----
__global__ void k_fuse_wmma(const float* __restrict__ xc,  // [B,256,169]
                            const float* __restrict__ Bm,  // [256][256]
                            const float* __restrict__ fb,  // [256]
                            const float* __restrict__ g, const float* __restrict__ bb,
                            const float* __restrict__ mn, const float* __restrict__ vr,
                            float* __restrict__ out) {     // [B,256,169]
  const int hw = 169, tiles = 11;
  const int b = blockIdx.x / tiles;
  const int pix0 = (blockIdx.x - b * tiles) << 4;
  const int tid = threadIdx.x;
  const int lane = tid & 31, wv = tid >> 5;
  const int half = lane >> 4, l16 = lane & 15;

  __shared__ float aLds[16 * 65];

  v8f acc[4];
#pragma unroll
  for (int j = 0; j < 4; ++j) acc[j] = (v8f){0.f,0.f,0.f,0.f,0.f,0.f,0.f,0.f};
  const int nbase = wv * 64;

  for (int kblk = 0; kblk < 4; ++kblk) {
    for (int e = tid; e < 16 * 64; e += 128) {
      int mr = e >> 6, kl = e & 63;
      int c = kblk * 64 + kl;
      int pix = pix0 + mr;
      aLds[mr * 65 + kl] =
          (pix < hw) ? xc[((size_t)b * 256 + c) * hw + pix] : 0.0f;
    }
    __syncthreads();

    const float* Bk = Bm + (size_t)(kblk * 64) * 256;
    for (int k0 = 0; k0 < 64; k0 += 4) {
      int kb = k0 + (half << 1);
      v2f a;
      a[0] = aLds[l16 * 65 + kb];
      a[1] = aLds[l16 * 65 + kb + 1];
      const float* Br0 = Bk + (size_t)kb * 256 + nbase + l16;
      const float* Br1 = Br0 + 256;
#pragma unroll
      for (int j = 0; j < 4; ++j) {
        v2f bf;
        bf[0] = Br0[j * 16];
        bf[1] = Br1[j * 16];
        acc[j] = wmma4(a, bf, acc[j]);
      }
    }
    __syncthreads();
  }

#pragma unroll
  for (int j = 0; j < 4; ++j) {
    int n = nbase + j * 16 + l16;
    float scale = g[n] * rsqrtf(vr[n] + BN_EPS);
    float shift = bb[n] - mn[n] * scale;
    float bs = fb[n];
#pragma unroll
    for (int r = 0; r < 8; ++r) {
      int pix = pix0 + r + (half << 3);
      if (pix < hw) {
        float v = (acc[j][r] + bs) * scale + shift;
        out[((size_t)b * 256 + n) * hw + pix] = fmaxf(v, 0.0f);
      }
    }
  }
}

// --- tiny cls (N=4) / box (N=16) heads --------------------------------------
__global__ void k_pred(const float* __restrict__ fc, const float* __restrict__ fr,
                       const float* __restrict__ cw, const float* __restrict__ cb,
                       const float* __restrict__ bw, const float* __restrict__ bbias,
                       float* __restrict__ out) {
  const int HW = 169;
  int idx = blockIdx.x * blockDim.x + threadIdx.x;
  if (idx < 16 * 4 * HW) {
    int b = idx / (4 * HW);
    int rem = idx - b * 4 * HW;
    int o = rem / HW, pix = rem - o * HW;
    float acc = cb[o];
    const float* f = fc + (size_t)b * 256 * HW + pix;
    const float* wr = cw + o * 256;
    for (int c = 0; c < 256; ++c) acc = fmaf(f[(size_t)c * HW], wr[c], acc);
    out[idx] = acc;
  } else {
    int j = idx - 16 * 4 * HW;
    if (j >= 16 * 16 * HW) return;
    int b = j / (16 * HW);
    int rem = j - b * 16 * HW;
    int o = rem / HW, pix = rem - o * HW;
    float acc = bbias[o];
    const float* f = fr + (size_t)b * 256 * HW + pix;
    const float* wr = bw + o * 256;
    for (int c = 0; c < 256; ++c) acc = fmaf(f[(size_t)c * HW], wr[c], acc);
    out[16 * 4 * HW + j] = acc;
  }
}

// ---------------------------------------------------------------------------
extern "C" void kernel_launch(void* const* d_in, const int* in_sizes, int n_in,
                              void* d_out, int out_size, void* d_ws, size_t ws_size,
                              hipStream_t stream) {
  (void)in_sizes; (void)n_in; (void)out_size;
  const float* exam  = (const float*)d_in[0];   // [16,256,20,20]
  const float* inst  = (const float*)d_in[1];   // [16,256,32,32]
  const float* off_w = (const float*)d_in[2];   // [4][27][2304]
  const float* off_b = (const float*)d_in[3];   // [4][27]
  const float* dcn_w = (const float*)d_in[4];   // [4][256][2304]
  const float* dcn_b = (const float*)d_in[5];   // [4][256]
  const float* bn1_g = (const float*)d_in[6];
  const float* bn1_b = (const float*)d_in[7];
  const float* bn1_m = (const float*)d_in[8];
  const float* bn1_v = (const float*)d_in[9];
  const float* fus_w = (const float*)d_in[10];  // [2][256][256]
  const float* fus_b = (const float*)d_in[11];
  const float* bn2_g = (const float*)d_in[12];
  const float* bn2_b = (const float*)d_in[13];
  const float* bn2_m = (const float*)d_in[14];
  const float* bn2_v = (const float*)d_in[15];
  const float* cls_w = (const float*)d_in[16];
  const float* cls_b = (const float*)d_in[17];
  const float* box_w = (const float*)d_in[18];
  const float* box_b = (const float*)d_in[19];

  float* ws = (float*)d_ws;
  // workspace layout (float offsets)
  const size_t off_om   = 0;                           // 16*27*900   = 388800
  const size_t off_tw   = off_om + 388800;             // float4 taps = 518400
  const size_t off_to   = off_tw + 518400;             // int4 taps   = 518400
  const size_t off_Boff = off_to + 518400;             // 2304*32     = 73728
  const size_t off_Bdcn = off_Boff + 73728;            // 2304*256    = 589824
  const size_t off_Bfus = off_Bdcn + 589824;           // 256*256     = 65536
  const size_t off_h0   = off_Bfus + 65536;            // 16*256*324
  const size_t off_h1   = off_h0 + 1327104;            // 16*256*900
  const size_t off_h2   = off_h1 + 3686400;
  const size_t off_h3   = off_h2 + 1327104;
  const size_t off_xc0  = off_h3 + 3686400;            // 16*256*169
  const size_t off_xc1  = off_xc0 + 692224;
  const size_t off_f0   = off_xc1 + 692224;
  const size_t off_f1   = off_f0 + 692224;
  const size_t total_f  = off_f1 + 692224;             // ~59.8 MB
  if (ws_size < total_f * sizeof(float)) return;

  float* hptr[4] = {ws + off_h0, ws + off_h1, ws + off_h2, ws + off_h3};

  for (int i = 0; i < 4; ++i) {
    const float* x = (i & 1) ? inst : exam;
    const int Hin = (i & 1) ? 32 : 20;
    const int Win = Hin, ho = Hin - 2, wo = ho;
    const int hw = ho * wo;
    const int tiles = (hw + 15) / 16;

    int tot = 2304 * 32;
    k_transpose<<<(tot + 255) / 256, 256, 0, stream>>>(
        off_w + (size_t)i * 27 * 2304, ws + off_Boff, 27, 2304, 32);

    k_offconv_wmma<<<16 * tiles, 32, 0, stream>>>(
        x, ws + off_Boff, off_b + i * 27, ws + off_om, Hin, Win, ho, wo);

    tot = 16 * 9 * hw;
    k_interp<<<(tot + 255) / 256, 256, 0, stream>>>(
        ws + off_om, (int4*)(ws + off_to), (float4*)(ws + off_tw), Hin, Win, ho, wo);

    tot = 2304 * 256;
    k_transpose<<<(tot + 255) / 256, 256, 0, stream>>>(
        dcn_w + (size_t)i * 256 * 2304, ws + off_Bdcn, 256, 2304, 256);

    k_dcn_wmma<<<16 * tiles, 128, 0, stream>>>(
        x, (const int4*)(ws + off_to), (const float4*)(ws + off_tw),
        ws + off_Bdcn, dcn_b + i * 256, bn1_g + i * 256, bn1_b + i * 256,
        bn1_m + i * 256, bn1_v + i * 256, hptr[i], Hin, Win, ho, wo);
  }

  k_xcorr<<<16 * 256, 256, 0, stream>>>(hptr[1], hptr[0], ws + off_xc0);
  k_xcorr<<<16 * 256, 256, 0, stream>>>(hptr[3], hptr[2], ws + off_xc1);

  for (int j = 0; j < 2; ++j) {
    int tot = 256 * 256;
    k_transpose<<<(tot + 255) / 256, 256, 0, stream>>>(
        fus_w + (size_t)j * 65536, ws + off_Bfus, 256, 256, 256);
    k_fuse_wmma<<<16 * 11, 128, 0, stream>>>(
        ws + (j ? off_xc1 : off_xc0), ws + off_Bfus, fus_b + j * 256,
        bn2_g + j * 256, bn2_b + j * 256, bn2_m + j * 256, bn2_v + j * 256,
        ws + (j ? off_f1 : off_f0));
  }

  k_pred<<<(54080 + 255) / 256, 256, 0, stream>>>(
      ws + off_f0, ws + off_f1, cls_w, cls_b, box_w, box_b, (float*)d_out);
}